// BoundaryAwareModule_1168231104864
// MI455X (gfx1250) — compile-verified
//
#include <hip/hip_runtime.h>
#include <math.h>

typedef float v2f __attribute__((ext_vector_type(2)));
typedef float v8f __attribute__((ext_vector_type(8)));

#define EPS 1e-5f
#define KNN 16
#define NT  4            // n-tiles (16 cols each) per wave in the WMMA GEMM

// ---------------------------------------------------------------------------
// Kernel 1: fused KNN (top-16 of pairwise sq-dist) + spatial stats
//   outputs: idx_out [B,N,16] (int), spatial [B,4,N] (mean_rel xyz, mean_d)
// One thread per query point; candidates streamed through LDS in 256 tiles.
// ---------------------------------------------------------------------------
__global__ void knn_spatial_kernel(const float* __restrict__ xyz,
                                   int* __restrict__ idx_out,
                                   float* __restrict__ spatial,
                                   int NB)
{
    __shared__ float s[256 * 3];
    const int chunkPerB = NB / 256;
    const int b   = blockIdx.x / chunkPerB;
    const int n   = (blockIdx.x % chunkPerB) * 256 + threadIdx.x;
    const float* xb = xyz + (size_t)b * NB * 3;

    const float qx = xb[n * 3 + 0];
    const float qy = xb[n * 3 + 1];
    const float qz = xb[n * 3 + 2];

    float bd[KNN];
    int   bi[KNN];
#pragma unroll
    for (int j = 0; j < KNN; ++j) { bd[j] = 3.4e38f; bi[j] = 0; }
    float worst = 3.4e38f;
    int   wslot = 0;

    for (int tile = 0; tile < NB; tile += 256) {
        __syncthreads();
        {
            const int ci = tile + threadIdx.x;
            s[threadIdx.x * 3 + 0] = xb[ci * 3 + 0];
            s[threadIdx.x * 3 + 1] = xb[ci * 3 + 1];
            s[threadIdx.x * 3 + 2] = xb[ci * 3 + 2];
        }
        __syncthreads();
        for (int t = 0; t < 256; ++t) {
            const float dx = s[t * 3 + 0] - qx;
            const float dy = s[t * 3 + 1] - qy;
            const float dz = s[t * 3 + 2] - qz;
            const float d  = fmaf(dx, dx, fmaf(dy, dy, dz * dz));
            if (d < worst) {
                const int ci2 = tile + t;
#pragma unroll
                for (int j = 0; j < KNN; ++j) {
                    const bool sel = (j == wslot);
                    bd[j] = sel ? d   : bd[j];
                    bi[j] = sel ? ci2 : bi[j];
                }
                worst = bd[0]; wslot = 0;
#pragma unroll
                for (int j = 1; j < KNN; ++j) {
                    if (bd[j] > worst) { worst = bd[j]; wslot = j; }
                }
            }
        }
    }

    // spatial stats over the 16 NN (self is always included with d=0)
    float sx = 0.f, sy = 0.f, sz = 0.f, sd = 0.f;
#pragma unroll
    for (int j = 0; j < KNN; ++j) {
        const int id = bi[j];
        const float rx = xb[id * 3 + 0] - qx;
        const float ry = xb[id * 3 + 1] - qy;
        const float rz = xb[id * 3 + 2] - qz;
        sx += rx; sy += ry; sz += rz;
        sd += sqrtf(fmaf(rx, rx, fmaf(ry, ry, rz * rz)));
        idx_out[((size_t)b * NB + n) * KNN + j] = id;
    }
    const float inv16 = 1.0f / 16.0f;
    spatial[((size_t)b * 4 + 0) * NB + n] = sx * inv16;
    spatial[((size_t)b * 4 + 1) * NB + n] = sy * inv16;
    spatial[((size_t)b * 4 + 2) * NB + n] = sz * inv16;
    spatial[((size_t)b * 4 + 3) * NB + n] = sd * inv16;
}

// ---------------------------------------------------------------------------
// Kernel 2: spatial-relation MLP  4 -> 32 (bn,relu) -> 64.  Weights in LDS.
// One thread per point.
// ---------------------------------------------------------------------------
__global__ void spatial_mlp_kernel(const float* __restrict__ spatial,
                                   const float* __restrict__ sw1, const float* __restrict__ sb1,
                                   const float* __restrict__ sg,  const float* __restrict__ sbe,
                                   const float* __restrict__ sm,  const float* __restrict__ sv,
                                   const float* __restrict__ sw2, const float* __restrict__ sb2,
                                   float* __restrict__ sfeat, int NB)
{
    __shared__ float w1[128], fb1[32], finv[32], w2[2048], b2[64];
    const int tid = threadIdx.x;
    for (int i = tid; i < 128;  i += 256) w1[i] = sw1[i];
    for (int i = tid; i < 2048; i += 256) w2[i] = sw2[i];
    if (tid < 64) b2[tid] = sb2[tid];
    if (tid < 32) {
        const float inv = sg[tid] * rsqrtf(sv[tid] + EPS);
        finv[tid] = inv;
        fb1[tid]  = sb1[tid] * inv + (sbe[tid] - sm[tid] * inv);
    }
    __syncthreads();

    const int g = blockIdx.x * 256 + tid;
    const int b = g / NB;
    const int n = g % NB;
    const float* sp = spatial + (size_t)b * 4 * NB + n;
    const float in0 = sp[0], in1 = sp[NB], in2 = sp[2 * (size_t)NB], in3 = sp[3 * (size_t)NB];

    float h[32];
#pragma unroll
    for (int o = 0; o < 32; ++o) {
        float a = w1[o * 4 + 0] * in0 + w1[o * 4 + 1] * in1
                + w1[o * 4 + 2] * in2 + w1[o * 4 + 3] * in3;
        h[o] = fmaxf(fmaf(a, finv[o], fb1[o]), 0.f);
    }
    float* outp = sfeat + (size_t)b * 64 * NB + n;
    for (int o = 0; o < 64; ++o) {
        float a = b2[o];
#pragma unroll
        for (int i = 0; i < 32; ++i) a = fmaf(w2[o * 32 + i], h[i], a);
        outp[(size_t)o * NB] = a;
    }
}

// ---------------------------------------------------------------------------
// Kernel 3: max_diff[b,c,n] = max_k( x[b,c,idx[n,k]] ) - x[b,c,n]
// One thread per (b,n); x rows are L2-resident so gathers hit L2.
// ---------------------------------------------------------------------------
__global__ void maxdiff_kernel(const float* __restrict__ x,
                               const int* __restrict__ idx,
                               float* __restrict__ out,
                               int C, int NB)
{
    const int g = blockIdx.x * blockDim.x + threadIdx.x;   // 0 .. B*N
    const int b = g / NB;
    const int n = g % NB;
    const float* xb = x + (size_t)b * C * NB;

    int nbr[KNN];
#pragma unroll
    for (int j = 0; j < KNN; ++j) nbr[j] = idx[(size_t)g * KNN + j];

    for (int c = 0; c < C; ++c) {
        const float* row = xb + (size_t)c * NB;
        const float self = row[n];
        float m = -3.4e38f;
#pragma unroll
        for (int j = 0; j < KNN; ++j) m = fmaxf(m, row[nbr[j]]);
        out[((size_t)b * C + c) * NB + n] = m - self;
    }
}

// ---------------------------------------------------------------------------
// Kernel 4: generic 1x1-conv GEMM via V_WMMA_F32_16X16X4_F32.
//   Y[b,m,n] = act( bn( sum_k W[m,k]*Xcat[b,k,n] + bias[m] ) )
//   Xcat = concat(X0[K0], X1[K1]) along channels.
// Block = 8 waves, all sharing one (b, m-tile); the 16 x Ktot weight tile is
// staged in LDS once per block (<=16KB). Each wave owns NT=4 n-tiles
// (16x64 output, 4 accumulators) so the LDS A-operand is amortized over 4
// WMMAs per k-step. Grids divide exactly -> no divergent guard, EXEC all-1s.
// act: 0=none 1=relu 2=sigmoid.  bng==nullptr -> no BN.
// ---------------------------------------------------------------------------
__global__ void gemm_wmma_kernel(const float* __restrict__ W, const float* __restrict__ bias,
                                 const float* __restrict__ X0, int K0,
                                 const float* __restrict__ X1, int K1,
                                 float* __restrict__ Y, int M, int NB, int Batch,
                                 const float* __restrict__ bng, const float* __restrict__ bnb,
                                 const float* __restrict__ bnm, const float* __restrict__ bnv,
                                 int act)
{
    __shared__ float s_w[16 * 256];          // column-major: s_w[k*16 + r]
    const int Ktot = K0 + K1;
    const int mt   = M >> 4;
    const int grpPerBM = (NB / 16) / (8 * NT);   // blocks per (b, m-tile)
    const int bm   = blockIdx.x / grpPerBM;
    const int ngrp = blockIdx.x % grpPerBM;
    const int b    = bm / mt;
    const int m0   = (bm % mt) * 16;

    // stage weight tile: rows m0..m0+15, all Ktot cols, column-major in LDS
    for (int i = threadIdx.x; i < 16 * Ktot; i += 256) {
        const int r = i & 15;
        const int k = i >> 4;
        s_w[i] = W[(size_t)(m0 + r) * Ktot + k];
    }
    __syncthreads();

    const int waveid = threadIdx.x >> 5;
    const int lane   = threadIdx.x & 31;
    const int lm     = lane & 15;        // M row (A) / N col (B)
    const int hi     = lane >> 4;        // lane half -> K pair {0,1} vs {2,3}
    const int koff   = hi * 2;
    const int n0     = (ngrp * 8 + waveid) * (16 * NT);

    v8f acc[NT];
#pragma unroll
    for (int t = 0; t < NT; ++t) acc[t] = (v8f){};

    // --- K loop over X0 ---
    {
        const float* Xb = X0 + (size_t)b * K0 * NB + n0 + lm;
        for (int k = 0; k < K0; k += 4) {
            v2f a;
            a.x = s_w[(k + koff) * 16 + lm];
            a.y = s_w[(k + koff + 1) * 16 + lm];
#pragma unroll
            for (int t = 0; t < NT; ++t) {
                v2f bv;
                bv.x = Xb[(size_t)(k + koff) * NB + 16 * t];
                bv.y = Xb[(size_t)(k + koff + 1) * NB + 16 * t];
                acc[t] = __builtin_amdgcn_wmma_f32_16x16x4_f32(
                    false, a, false, bv, (short)0, acc[t], false, false);
            }
        }
    }
    // --- K loop over X1 (concatenated channels) ---
    if (X1 != nullptr) {
        const float* Xb = X1 + (size_t)b * K1 * NB + n0 + lm;
        const float* sw1p = s_w + K0 * 16;
        for (int k = 0; k < K1; k += 4) {
            v2f a;
            a.x = sw1p[(k + koff) * 16 + lm];
            a.y = sw1p[(k + koff + 1) * 16 + lm];
#pragma unroll
            for (int t = 0; t < NT; ++t) {
                v2f bv;
                bv.x = Xb[(size_t)(k + koff) * NB + 16 * t];
                bv.y = Xb[(size_t)(k + koff + 1) * NB + 16 * t];
                acc[t] = __builtin_amdgcn_wmma_f32_16x16x4_f32(
                    false, a, false, bv, (short)0, acc[t], false, false);
            }
        }
    }

    // --- epilogue: bias + BN + activation, per C/D layout (vgpr r -> M=r(+8)) ---
#pragma unroll
    for (int r = 0; r < 8; ++r) {
        const int ch = m0 + r + hi * 8;
        float scale = 1.0f, shift = bias[ch];
        if (bng != nullptr) {
            const float inv = bng[ch] * rsqrtf(bnv[ch] + EPS);
            shift = shift * inv + (bnb[ch] - bnm[ch] * inv);
            scale = inv;
        }
        float* Yp = Y + ((size_t)b * M + ch) * NB + n0 + lm;
#pragma unroll
        for (int t = 0; t < NT; ++t) {
            float v = fmaf(acc[t][r], scale, shift);
            if (act == 1)      v = fmaxf(v, 0.f);
            else if (act == 2) v = 1.0f / (1.0f + __expf(-v));
            Yp[16 * t] = v;
        }
    }
}

// ---------------------------------------------------------------------------
// Kernel 5: out = x + boundary_feat * attn   (attn already sigmoid'ed)
// ---------------------------------------------------------------------------
__global__ void combine_kernel(const float* __restrict__ x,
                               const float* __restrict__ bfeat,
                               const float* __restrict__ attn,
                               float* __restrict__ out, int total)
{
    const int g = blockIdx.x * blockDim.x + threadIdx.x;
    if (g < total) out[g] = fmaf(bfeat[g], attn[g], x[g]);
}

// ---------------------------------------------------------------------------
extern "C" void kernel_launch(void* const* d_in, const int* in_sizes, int n_in,
                              void* d_out, int out_size, void* d_ws, size_t ws_size,
                              hipStream_t stream)
{
    (void)in_sizes; (void)n_in; (void)out_size; (void)ws_size;
    const int B = 2, C = 128, N = 8192;

    const float* x    = (const float*)d_in[0];
    const float* xyz  = (const float*)d_in[1];
    const float* bw1  = (const float*)d_in[2];
    const float* bb1  = (const float*)d_in[3];
    const float* bn1g = (const float*)d_in[4];
    const float* bn1b = (const float*)d_in[5];
    const float* bn1m = (const float*)d_in[6];
    const float* bn1v = (const float*)d_in[7];
    const float* bw2  = (const float*)d_in[8];
    const float* bb2  = (const float*)d_in[9];
    const float* bn2g = (const float*)d_in[10];
    const float* bn2b = (const float*)d_in[11];
    const float* bn2m = (const float*)d_in[12];
    const float* bn2v = (const float*)d_in[13];
    const float* sw1  = (const float*)d_in[14];
    const float* sb1  = (const float*)d_in[15];
    const float* sbng = (const float*)d_in[16];
    const float* sbnb = (const float*)d_in[17];
    const float* sbnm = (const float*)d_in[18];
    const float* sbnv = (const float*)d_in[19];
    const float* sw2  = (const float*)d_in[20];
    const float* sb2  = (const float*)d_in[21];
    const float* aw1  = (const float*)d_in[22];
    const float* ab1  = (const float*)d_in[23];
    const float* abng = (const float*)d_in[24];
    const float* abnb = (const float*)d_in[25];
    const float* abnm = (const float*)d_in[26];
    const float* abnv = (const float*)d_in[27];
    const float* aw2  = (const float*)d_in[28];
    const float* ab2  = (const float*)d_in[29];

    // workspace layout (bytes)
    char* ws = (char*)d_ws;
    size_t off = 0;
    int*   ws_idx   = (int*)  (ws + off); off += (size_t)B * N * KNN * 4;  // 1 MB
    float* ws_spat  = (float*)(ws + off); off += (size_t)B * 4 * N * 4;    // 256 KB
    float* ws_sfeat = (float*)(ws + off); off += (size_t)B * 64 * N * 4;   // 4 MB
    float* ws_mdiff = (float*)(ws + off); off += (size_t)B * C * N * 4;    // 8 MB
    float* ws_y1    = (float*)(ws + off); off += (size_t)B * C * N * 4;    // 8 MB
    float* ws_bfeat = (float*)(ws + off); off += (size_t)B * C * N * 4;    // 8 MB
    float* ws_a1    = (float*)(ws + off); off += (size_t)B * 64 * N * 4;   // 4 MB
    float* ws_attn  = (float*)(ws + off); off += (size_t)B * C * N * 4;    // 8 MB

    const int grpPerBM = (N / 16) / (8 * NT);   // 16 blocks per (b, m-tile)

    // 1) KNN + spatial stats
    knn_spatial_kernel<<<B * (N / 256), 256, 0, stream>>>(xyz, ws_idx, ws_spat, N);

    // 2) spatial MLP -> spatial_feat [B,64,N]
    spatial_mlp_kernel<<<(B * N) / 256, 256, 0, stream>>>(
        ws_spat, sw1, sb1, sbng, sbnb, sbnm, sbnv, sw2, sb2, ws_sfeat, N);

    // 3) max_diff [B,C,N]
    maxdiff_kernel<<<(B * N) / 256, 256, 0, stream>>>(x, ws_idx, ws_mdiff, C, N);

    // 4a) y1 = relu(bn1(bw1 . [x;max_diff] + bb1))   M=128, K=256
    gemm_wmma_kernel<<<B * (C / 16) * grpPerBM, 256, 0, stream>>>(
        bw1, bb1, x, C, ws_mdiff, C, ws_y1, C, N, B,
        bn1g, bn1b, bn1m, bn1v, 1);

    // 4b) boundary_feat = relu(bn2(bw2 . y1 + bb2))  M=128, K=128
    gemm_wmma_kernel<<<B * (C / 16) * grpPerBM, 256, 0, stream>>>(
        bw2, bb2, ws_y1, C, nullptr, 0, ws_bfeat, C, N, B,
        bn2g, bn2b, bn2m, bn2v, 1);

    // 4c) a1 = relu(abn(aw1 . [x;spatial_feat] + ab1))  M=64, K=192
    gemm_wmma_kernel<<<B * (64 / 16) * grpPerBM, 256, 0, stream>>>(
        aw1, ab1, x, C, ws_sfeat, 64, ws_a1, 64, N, B,
        abng, abnb, abnm, abnv, 1);

    // 4d) attn = sigmoid(aw2 . a1 + ab2)  M=128, K=64, no BN
    gemm_wmma_kernel<<<B * (C / 16) * grpPerBM, 256, 0, stream>>>(
        aw2, ab2, ws_a1, 64, nullptr, 0, ws_attn, C, N, B,
        nullptr, nullptr, nullptr, nullptr, 2);

    // 5) out = x + boundary_feat * attn
    {
        const int total = B * C * N;
        combine_kernel<<<(total + 255) / 256, 256, 0, stream>>>(
            x, ws_bfeat, ws_attn, (float*)d_out, total);
    }
}